// DeepHough_10831907521089
// MI455X (gfx1250) — compile-verified
//
#include <hip/hip_runtime.h>

// Deep Hough Transform as per-angle GEMM with register-synthesized one-hot B,
// using V_WMMA_F32_16X16X4_F32 (exact f32 path, matches f32 reference numerics).
//
// out[nc, a, r] = sum_p feat[nc, p] * (rho(a,p) == r)
//
// Each wave owns TWO 16-row M-tiles (32 rows of NC) x all 8 N-tiles (rho 0..127),
// so every register-built one-hot B tile feeds 16 WMMAs -> VALU:WMMA < 3:1.
// feat (67MB) is L2-resident (192MB L2), so the 120 angle passes stream from L2.
// Accumulators live in VGPRs across all 4096 K-steps: zero atomics, zero LDS.

typedef float v2f __attribute__((ext_vector_type(2)));
typedef float v8f __attribute__((ext_vector_type(8)));

#define DHT_H        128
#define DHT_W        128
#define DHT_HW       (DHT_H * DHT_W)     // 16384
#define DHT_NUMANGLE 120
#define DHT_NUMRHO   120
#define DHT_PI_D     3.14159265358979323846

// rho bin for pixel p at angle tables (tc, ts).
// Matches numpy: f32 mul, f32 mul, f32 add (NO fma contraction), round half-even.
__device__ __forceinline__ int dht_rho(int p, float tc, float ts) {
    int x = p & (DHT_W - 1);
    int y = p >> 7;                       // p / 128
    float fx = (float)(x - DHT_W / 2);
    float fy = (float)(y - DHT_H / 2);
    float v = __fadd_rn(__fmul_rn(fx, tc), __fmul_rn(fy, ts));
    int r = (int)rintf(v) + (DHT_NUMRHO / 2);
    r = r < 0 ? 0 : r;
    r = r > (DHT_NUMRHO - 1) ? (DHT_NUMRHO - 1) : r;
    return r;
}

__global__ __launch_bounds__(32)
void dht_wmma_f32(const float* __restrict__ feat, float* __restrict__ out) {
    const int mtile = blockIdx.x;          // which 32-row slice of NC=1024
    const int a     = blockIdx.y;          // angle
    const int lane  = threadIdx.x & 31;
    const int nlane = lane & 15;           // N index within tile / M row for A
    const int half  = lane >> 4;           // 0: K={0,1}, 1: K={2,3}
    const int koff  = half * 2;

    // Angle tables: float64 cos/sin then cast to f32, exactly like the reference.
    // irho = int(sqrt(128^2+128^2)+1)/(numrho-1) = 182/119.
    const double ang  = (double)a * (DHT_PI_D / (double)DHT_NUMANGLE);
    const double irho = 182.0 / 119.0;
    const float  tc   = (float)(cos(ang) / irho);
    const float  ts   = (float)(sin(ang) / irho);

    const int mbase = mtile * 32;
    // A-layout (16x4 f32): lane L -> M = L%16, K = 2*(L/16) + {0,1}  => float2 load.
    const float* arow0 = feat + (size_t)(mbase + nlane) * DHT_HW + koff;
    const float* arow1 = arow0 + (size_t)16 * DHT_HW;

    v8f acc0[8], acc1[8];
#pragma unroll
    for (int nt = 0; nt < 8; ++nt) {
        acc0[nt] = (v8f){0.f, 0.f, 0.f, 0.f, 0.f, 0.f, 0.f, 0.f};
        acc1[nt] = (v8f){0.f, 0.f, 0.f, 0.f, 0.f, 0.f, 0.f, 0.f};
    }

    for (int kb = 0; kb < DHT_HW; kb += 4) {
        v2f av0 = *(const v2f*)(arow0 + kb);
        v2f av1 = *(const v2f*)(arow1 + kb);

        // B-layout mirrors A: lane L holds B[K=koff+j, N=L%16] in vgpr j.
        // Same two pixels as this lane's A elements -> one rho pair feeds all tiles.
        const int p0 = kb + koff;
        const int r0 = dht_rho(p0, tc, ts);
        const int r1 = dht_rho(p0 + 1, tc, ts);

#pragma unroll
        for (int nt = 0; nt < 8; ++nt) {
            const int n = nt * 16 + nlane;
            v2f bv;
            bv.x = (r0 == n) ? 1.0f : 0.0f;
            bv.y = (r1 == n) ? 1.0f : 0.0f;
            // 8 args: (neg_a, A, neg_b, B, c_mod, C, reuse_a, reuse_b)
            acc0[nt] = __builtin_amdgcn_wmma_f32_16x16x4_f32(
                false, av0, false, bv, (short)0, acc0[nt], false, false);
            acc1[nt] = __builtin_amdgcn_wmma_f32_16x16x4_f32(
                false, av1, false, bv, (short)0, acc1[nt], false, false);
        }
    }

    // D-layout (16x16 f32): vgpr j, lane L -> M = j + 8*(L/16), N = L%16.
#pragma unroll
    for (int nt = 0; nt < 8; ++nt) {
        const int r = nt * 16 + nlane;
        if (r >= DHT_NUMRHO) continue;     // pad lanes (r=120..127) drop out
#pragma unroll
        for (int j = 0; j < 8; ++j) {
            const int row0 = mbase + j + 8 * half;        // nc index, tile 0
            out[(size_t)row0 * (DHT_NUMANGLE * DHT_NUMRHO) + a * DHT_NUMRHO + r] =
                acc0[nt][j];
            const int row1 = row0 + 16;                   // nc index, tile 1
            out[(size_t)row1 * (DHT_NUMANGLE * DHT_NUMRHO) + a * DHT_NUMRHO + r] =
                acc1[nt][j];
        }
    }
}

extern "C" void kernel_launch(void* const* d_in, const int* in_sizes, int n_in,
                              void* d_out, int out_size, void* d_ws, size_t ws_size,
                              hipStream_t stream) {
    const float* feat = (const float*)d_in[0];
    float* out = (float*)d_out;

    const int NC = in_sizes[0] / DHT_HW;   // 8*128 = 1024 rows
    dim3 grid(NC / 32, DHT_NUMANGLE);      // 32 dual-M-tiles x 120 angles
    dht_wmma_f32<<<grid, 32, 0, stream>>>(feat, out);
}